// PINN_75239237091852
// MI455X (gfx1250) — compile-verified
//
#include <hip/hip_runtime.h>
#include <math.h>

typedef __attribute__((ext_vector_type(2))) float v2f;
typedef __attribute__((ext_vector_type(8))) float v8f;

#define HDIM 128
#define NLAY 3
#define PPB 16              // points per block
#define NC 6                // jet components: v, gx, gy, hxx, hxy, hyy
#define NCOL (NC * PPB)     // 96 GEMM columns
#define CP 112              // padded col stride (per K-pair row): 2*CP % 64 == 32 -> bank-safe
#define WS 132              // padded row stride for W tile: 132 % 64 == 4 -> bank-safe
#define WGSIZE 256

// K-pair interleaved activation layout: element (k, col) lives at
//   (k>>1) * (2*CP) + 2*col + (k&1)
// so rows (2j, 2j+1) of one column are an 8-byte-aligned contiguous pair
// -> B fragments load with a single ds_load_b64 (no register repacking).
#define HIDX(k, col) ((((k) >> 1) * (2 * CP)) + (((col)) << 1) + ((k) & 1))

// ---------------------------------------------------------------------------
// One network pass: builds the 2nd-order jet of the MLP output for PPB points.
// ---------------------------------------------------------------------------
__device__ __forceinline__ void run_net(
    float* __restrict__ Wlds, float* __restrict__ Hl, float* __restrict__ sbuf,
    float* __restrict__ jet,
    const float* __restrict__ W1, const float* __restrict__ b1,
    const float* __restrict__ Wh, const float* __restrict__ bh,
    const float* __restrict__ Wf, const float* __restrict__ bf,
    const float* __restrict__ xs, const float* __restrict__ ys)
{
    const int t    = threadIdx.x;
    const int lane = t & 31;
    const int wave = t >> 5;

    // ---- input layer (2 -> 128): build initial jet directly ----
    {
        const int p = t & 15;
        const float x = xs[p], y = ys[p];
        #pragma unroll
        for (int j = 0; j < 8; ++j) {
            const int i = (t >> 4) + 16 * j;
            const float w0 = W1[i * 2 + 0], w1 = W1[i * 2 + 1];
            const float z  = fmaf(w0, x, fmaf(w1, y, b1[i]));
            const float tv = tanhf(z);
            const float t1 = 1.f - tv * tv;
            const float t2 = -2.f * tv * t1;
            Hl[HIDX(i, 0 * PPB + p)] = tv;
            Hl[HIDX(i, 1 * PPB + p)] = t1 * w0;
            Hl[HIDX(i, 2 * PPB + p)] = t1 * w1;
            Hl[HIDX(i, 3 * PPB + p)] = t2 * w0 * w0;
            Hl[HIDX(i, 4 * PPB + p)] = t2 * w0 * w1;
            Hl[HIDX(i, 5 * PPB + p)] = t2 * w1 * w1;
        }
    }
    __syncthreads();

    // ---- hidden layers: Z = W @ Hjet via V_WMMA_F32_16X16X4_F32 ----
    for (int l = 0; l < NLAY; ++l) {
        const float* __restrict__ W = Wh + l * HDIM * HDIM;

        // stage weights into LDS (float4 -> ds_store_b128), padded stride
        for (int idx = t; idx < HDIM * (HDIM / 4); idx += WGSIZE) {
            const int row = idx >> 5;
            const int q   = idx & 31;
            const float4 v = *(const float4*)(W + row * HDIM + q * 4);
            *(float4*)(Wlds + row * WS + q * 4) = v;
        }
        // warm L2 with next layer's weights while we compute
        if (l + 1 < NLAY) {
            __builtin_prefetch(Wh + (l + 1) * HDIM * HDIM + t * 64, 0, 1);
        }
        __syncthreads();

        const int rbase  = wave * 16;
        const int m      = lane & 15;
        const int khalf  = (lane < 16) ? 0 : 2;
        const int rowoff = (lane < 16) ? 0 : 8;

        v8f acc[NC];
        const v8f vzero = {0.f, 0.f, 0.f, 0.f, 0.f, 0.f, 0.f, 0.f};
        #pragma unroll
        for (int c = 0; c < NC; ++c) acc[c] = vzero;

        // K loop: 128 in steps of 4; A-frag (ds_load_b64) reused across 6 col tiles
        for (int kt = 0; kt < HDIM / 4; ++kt) {
            const int kb = kt * 4;
            const v2f a = *(const v2f*)&Wlds[(rbase + m) * WS + kb + khalf];
            #pragma unroll
            for (int c = 0; c < NC; ++c) {
                const v2f b = *(const v2f*)&Hl[HIDX(kb + khalf, c * PPB + m)];
                acc[c] = __builtin_amdgcn_wmma_f32_16x16x4_f32(
                    false, a, false, b, (short)0, acc[c], false, false);
            }
        }
        __syncthreads();   // all waves done reading Hl

        // tanh-jet nonlinearity applied straight from accumulators:
        // lane holds all 6 components of (row = rbase+rowoff+r, point = m)
        #pragma unroll
        for (int r = 0; r < 8; ++r) {
            const int i  = rbase + rowoff + r;
            const float zv  = acc[0][r] + bh[l * HDIM + i];
            const float zgx = acc[1][r];
            const float zgy = acc[2][r];
            const float zxx = acc[3][r];
            const float zxy = acc[4][r];
            const float zyy = acc[5][r];
            const float tv = tanhf(zv);
            const float t1 = 1.f - tv * tv;
            const float t2 = -2.f * tv * t1;
            Hl[HIDX(i, 0 * PPB + m)] = tv;
            Hl[HIDX(i, 1 * PPB + m)] = t1 * zgx;
            Hl[HIDX(i, 2 * PPB + m)] = t1 * zgy;
            Hl[HIDX(i, 3 * PPB + m)] = fmaf(t2, zgx * zgx, t1 * zxx);
            Hl[HIDX(i, 4 * PPB + m)] = fmaf(t2, zgx * zgy, t1 * zxy);
            Hl[HIDX(i, 5 * PPB + m)] = fmaf(t2, zgy * zgy, t1 * zyy);
        }
        __syncthreads();
    }

    // ---- final layer (128 -> 1): s[c][p] = sum_k Wf[k] * Hjet[k][c,p] ----
    if (t < 2 * NCOL) {
        const int col  = t % NCOL;
        const int half = t / NCOL;
        float s = 0.f;
        #pragma unroll 4
        for (int k = half * 64; k < half * 64 + 64; ++k)
            s = fmaf(Wf[k], Hl[HIDX(k, col)], s);
        sbuf[half * NCOL + col] = s;
    }
    __syncthreads();
    if (t < PPB) {
        const int p = t;
        const float sv  = sbuf[0 * PPB + p] + sbuf[NCOL + 0 * PPB + p] + bf[0];
        const float sgx = sbuf[1 * PPB + p] + sbuf[NCOL + 1 * PPB + p];
        const float sgy = sbuf[2 * PPB + p] + sbuf[NCOL + 2 * PPB + p];
        const float sxx = sbuf[3 * PPB + p] + sbuf[NCOL + 3 * PPB + p];
        const float sxy = sbuf[4 * PPB + p] + sbuf[NCOL + 4 * PPB + p];
        const float syy = sbuf[5 * PPB + p] + sbuf[NCOL + 5 * PPB + p];
        const float tv = tanhf(sv);
        const float t1 = 1.f - tv * tv;
        const float t2 = -2.f * tv * t1;
        jet[0 * PPB + p] = tv;
        jet[1 * PPB + p] = t1 * sgx;
        jet[2 * PPB + p] = t1 * sgy;
        jet[3 * PPB + p] = fmaf(t2, sgx * sgx, t1 * sxx);
        jet[4 * PPB + p] = fmaf(t2, sgx * sgy, t1 * sxy);
        jet[5 * PPB + p] = fmaf(t2, sgy * sgy, t1 * syy);
    }
    __syncthreads();
}

__global__ void __launch_bounds__(WGSIZE)
pinn_elasticity_kernel(
    const float* __restrict__ x, const float* __restrict__ y,
    const float* __restrict__ W1u, const float* __restrict__ b1u,
    const float* __restrict__ Whu, const float* __restrict__ bhu,
    const float* __restrict__ Wfu, const float* __restrict__ bfu,
    const float* __restrict__ W1v, const float* __restrict__ b1v,
    const float* __restrict__ Whv, const float* __restrict__ bhv,
    const float* __restrict__ Wfv, const float* __restrict__ bfv,
    const float* __restrict__ mu, const float* __restrict__ lmbd,
    float* __restrict__ out, int N)
{
    extern __shared__ float lds[];
    float* Wlds = lds;                       // HDIM*WS   = 16896 floats
    float* Hl   = Wlds + HDIM * WS;          // 64 pair-rows * 2*CP = 14336 floats
    float* sbuf = Hl + HDIM * CP;            // 2*NCOL    =   192 floats
    float* jets = sbuf + 2 * NCOL;           // 2*NC*PPB  =   192 floats
    float* xs   = jets + 2 * NC * PPB;       // PPB
    float* ys   = xs + PPB;                  // PPB

    const int t  = threadIdx.x;
    const int p0 = blockIdx.x * PPB;

    if (t < PPB) {
        const int gp = p0 + t;
        xs[t] = (gp < N) ? x[gp] : 0.f;
        ys[t] = (gp < N) ? y[gp] : 0.f;
    }
    __syncthreads();

    run_net(Wlds, Hl, sbuf, jets,            W1u, b1u, Whu, bhu, Wfu, bfu, xs, ys);
    run_net(Wlds, Hl, sbuf, jets + NC * PPB, W1v, b1v, Whv, bhv, Wfv, bfv, xs, ys);

    if (t < PPB) {
        const int p  = t;
        const int gp = p0 + p;
        if (gp < N) {
            const float* ju = jets;
            const float* jv = jets + NC * PPB;
            const float Uv  = ju[0 * PPB + p];
            const float Uxx = ju[3 * PPB + p];
            const float Uxy = ju[4 * PPB + p];
            const float Uyy = ju[5 * PPB + p];
            const float Vv  = jv[0 * PPB + p];
            const float Vxx = jv[3 * PPB + p];
            const float Vxy = jv[4 * PPB + p];
            const float Vyy = jv[5 * PPB + p];

            const float MU = mu[0], LA = lmbd[0];
            const float C11 = 2.f * MU + LA;
            const float C12 = LA;
            const float C33 = 2.f * MU;

            // rx = d(Sxx)/dx + d(Sxy)/dy ; ry = d(Sxy)/dx + d(Syy)/dy
            const float rx = C11 * Uxx + C12 * Vxy + C33 * (Uyy + 0.5f * Vxy);
            const float ry = C33 * (Uxy + 0.5f * Vxx) + C11 * Vyy + C12 * Uxy;

            const float PI = 3.14159265358979323846f;
            const float Qc = 4.0f;
            const float xx = xs[p], yy = ys[p];
            const float y2 = yy * yy, y3 = y2 * yy, y4 = y3 * yy;
            const float s2px = sinf(2.f * PI * xx), c2px = cosf(2.f * PI * xx);
            const float spx  = sinf(PI * xx),       cpx  = cosf(PI * xx);
            const float spy  = sinf(PI * yy),       cpy  = cosf(PI * yy);
            const float PI2  = PI * PI;

            const float Fx = -1.0f * (4.f * PI2 * c2px * spy - Qc * y3 * PI * cpx)
                             - 0.5f * (PI2 * c2px * spy - Qc * y3 * PI * cpx)
                             - 0.5f * 8.f * PI2 * c2px * spy;
            const float Fy =  1.0f * (3.f * Qc * y2 * spx - 2.f * PI2 * cpy * s2px)
                             - 0.5f * (2.f * PI2 * cpy * s2px + Qc * y4 * PI2 * spx * 0.25f)
                             + 0.5f * 6.f * Qc * y2 * spx;

            out[0 * N + gp] = rx - Fx;   // Lx
            out[1 * N + gp] = ry - Fy;   // Ly
            out[2 * N + gp] = Uv;        // U
            out[3 * N + gp] = Vv;        // V
        }
    }
}

extern "C" void kernel_launch(void* const* d_in, const int* in_sizes, int n_in,
                              void* d_out, int out_size, void* d_ws, size_t ws_size,
                              hipStream_t stream) {
    const float* x    = (const float*)d_in[0];
    const float* y    = (const float*)d_in[1];
    const float* W1u  = (const float*)d_in[2];
    const float* b1u  = (const float*)d_in[3];
    const float* Whu  = (const float*)d_in[4];
    const float* bhu  = (const float*)d_in[5];
    const float* Wfu  = (const float*)d_in[6];
    const float* bfu  = (const float*)d_in[7];
    const float* W1v  = (const float*)d_in[8];
    const float* b1v  = (const float*)d_in[9];
    const float* Whv  = (const float*)d_in[10];
    const float* bhv  = (const float*)d_in[11];
    const float* Wfv  = (const float*)d_in[12];
    const float* bfv  = (const float*)d_in[13];
    const float* mu   = (const float*)d_in[14];
    const float* lmbd = (const float*)d_in[15];

    const int N = in_sizes[0];
    const int grid = (N + PPB - 1) / PPB;
    const size_t shmem =
        (size_t)(HDIM * WS + HDIM * CP + 2 * NCOL + 2 * NC * PPB + 2 * PPB) * sizeof(float);

    pinn_elasticity_kernel<<<grid, WGSIZE, shmem, stream>>>(
        x, y, W1u, b1u, Whu, bhu, Wfu, bfu,
        W1v, b1v, Whv, bhv, Wfv, bfv, mu, lmbd,
        (float*)d_out, N);
}